// GCN_52115133170059
// MI455X (gfx1250) — compile-verified
//
#include <hip/hip_runtime.h>
#include <hip/hip_bf16.h>
#include <math.h>

typedef __attribute__((ext_vector_type(2))) float v2f;
typedef __attribute__((ext_vector_type(8))) float v8f;

// ---------------------------------------------------------------------------
// GEMM: Y[N x ncols] = X[N x 128] @ W[128 x ncols], fp32 via V_WMMA_F32_16X16X4_F32
//
// Block: 256 threads = 8 waves; covers a 64-row strip of X (4 row-tiles).
// Wave w owns the 16-wide column tile [16w, 16w+16) for ALL 4 row tiles,
// so each B fragment is reused by 4 WMMAs (4:5 WMMA:LDS-load ratio) and the
// W panel staging cost is amortized over 4x more matrix work.
//
// LDS staging:
//   Xs[64][132] : X strip, row-major, padded stride (bank-conflict free)
//   Wt[128][36] : one 32-wide K-panel of W, TRANSPOSED (Wt[col][k]),
//                 zero-padded for col >= ncols -> no predication in hot loop.
//
// Fragment layouts (ISA 7.12.2, 32-bit operands):
//   A 16x4 : lanes 0-15 hold {K=k,k+1} for M=lane; lanes 16-31 hold {K=k+2,k+3}
//   B 4x16 : symmetric (lane<16: {K=k,k+1} at N=lane; lane>=16: {K=k+2,k+3})
//   C/D    : lane<16 -> M=r, lane>=16 -> M=8+r; N = lane&15
// ---------------------------------------------------------------------------
__global__ __launch_bounds__(256)
void gemm_wmma_f32(const float* __restrict__ X, const float* __restrict__ W,
                   float* __restrict__ Y, int Nrows, int ncols) {
    constexpr int K  = 128;
    constexpr int XS = K + 4;       // 132
    constexpr int KP = 32;          // K panel height
    constexpr int WT = KP + 4;      // 36
    constexpr int RT = 4;           // row tiles per block
    constexpr int BM = RT * 16;     // 64 rows per block

    __shared__ float Xs[BM * XS];   // 33792 B
    __shared__ float Wt[128 * WT];  // 18432 B

    const int row0 = blockIdx.x * BM;
    const int tid  = threadIdx.x;

    // ---- stage X strip (coalesced global, zero-pad OOB rows), 32-bit math ----
    #pragma unroll
    for (int i = tid; i < BM * K; i += 256) {
        const int r  = i >> 7;
        const int c  = i & (K - 1);
        const int gr = row0 + r;
        Xs[r * XS + c] = (gr < Nrows) ? X[(unsigned)gr * (unsigned)K + (unsigned)c] : 0.0f;
    }

    const int wave  = tid >> 5;     // column tile 0..7
    const int lane  = tid & 31;
    const int laneM = lane & 15;
    const int half  = lane >> 4;
    const int col0  = wave * 16;
    const int col   = col0 + laneM;
    const bool waveActive = (col0 < ncols);   // wave-uniform
    const bool colOK      = (col  < ncols);

    const float* aBase = &Xs[laneM * XS + 2 * half];
    const float* bBase = &Wt[col  * WT + 2 * half];

    v8f acc[RT];
    #pragma unroll
    for (int rt = 0; rt < RT; ++rt)
        acc[rt] = (v8f){0.f, 0.f, 0.f, 0.f, 0.f, 0.f, 0.f, 0.f};

    for (int p = 0; p < K / KP; ++p) {
        __syncthreads();   // covers Xs fill (p==0) / prior panel reads (p>0)

        // ---- stage transposed, zero-padded W panel: Wt[c][r] = W[p*32+r][c] ----
        #pragma unroll
        for (int i = tid; i < KP * 128; i += 256) {
            const int r = i >> 7;      // k within panel
            const int c = i & 127;     // column
            const float v = (c < ncols)
                ? W[(unsigned)(p * KP + r) * (unsigned)ncols + (unsigned)c] : 0.0f;
            Wt[c * WT + r] = v;
        }
        __syncthreads();

        if (waveActive) {
            const float* aP = aBase + p * KP;
            #pragma unroll
            for (int k = 0; k < KP; k += 4) {
                const float2 b = *(const float2*)(bBase + k);      // ds_load_b64
                const v2f bv = {b.x, b.y};
                #pragma unroll
                for (int rt = 0; rt < RT; ++rt) {
                    const float2 a = *(const float2*)(aP + rt * 16 * XS + k);
                    const v2f av = {a.x, a.y};
                    acc[rt] = __builtin_amdgcn_wmma_f32_16x16x4_f32(
                        false, av, false, bv, (short)0, acc[rt], false, false);
                }
            }
        }
    }

    if (waveActive) {
        #pragma unroll
        for (int rt = 0; rt < RT; ++rt) {
            #pragma unroll
            for (int r = 0; r < 8; ++r) {
                const int gr = row0 + rt * 16 + (half ? (8 + r) : r);
                if (gr < Nrows && colOK) {
                    Y[(unsigned)gr * (unsigned)ncols + (unsigned)col] = acc[rt][r];
                }
            }
        }
    }
}

// ---------------------------------------------------------------------------
// Zero-fill
// ---------------------------------------------------------------------------
__global__ void zero_f32(float* __restrict__ p, unsigned n) {
    const unsigned i = blockIdx.x * blockDim.x + threadIdx.x;
    if (i < n) p[i] = 0.0f;
}

// ---------------------------------------------------------------------------
// Edge aggregation: out[dst[e]][:] += xw[src[e]][:]
// One thread per (edge, 4-float chunk); CHUNKS compile-time so /,% are cheap.
// Gathers are L2-resident (feature matrix <= 51.2 MB << 192 MB L2).
// ---------------------------------------------------------------------------
template <int CHUNKS>
__global__ __launch_bounds__(256)
void agg_edges(const float* __restrict__ xw, const int* __restrict__ src,
               const int* __restrict__ dst, float* __restrict__ out, int E) {
    constexpr int F = CHUNKS * 4;
    const unsigned gid = blockIdx.x * blockDim.x + threadIdx.x;
    if (gid >= (unsigned)E * (unsigned)CHUNKS) return;
    const int e = (int)(gid / CHUNKS);
    const int c = (int)(gid % CHUNKS);

    if (c == 0) {                        // speculative look-ahead on edge lists
        const int pe = e + 8192;
        if (pe < E) {
            __builtin_prefetch(&src[pe], 0, 0);   // -> global_prefetch_b8
            __builtin_prefetch(&dst[pe], 0, 0);
        }
    }

    const int s = src[e];
    const int d = dst[e];
    const float4 v = *(const float4*)(xw + (unsigned)s * F + (unsigned)c * 4u);
    float* o = out + (unsigned)d * F + (unsigned)c * 4u;
    atomicAdd(o + 0, v.x);
    atomicAdd(o + 1, v.y);
    atomicAdd(o + 2, v.z);
    atomicAdd(o + 3, v.w);
}

// ---------------------------------------------------------------------------
// Fused bias + BatchNorm(eval) + ReLU, in place:
//   h = relu((h + b - rm) * g * rsqrt(rv+eps) + beta)
// ---------------------------------------------------------------------------
__global__ void bias_bn_relu(float* __restrict__ h, const float* __restrict__ b,
                             const float* __restrict__ g, const float* __restrict__ beta,
                             const float* __restrict__ rm, const float* __restrict__ rv,
                             unsigned total, int F) {
    const unsigned i = blockIdx.x * blockDim.x + threadIdx.x;
    if (i >= total) return;
    const int c = (int)(i & (unsigned)(F - 1));   // F = 128 (power of two)
    const float s  = g[c] * rsqrtf(rv[c] + 1e-5f);
    const float sh = (b[c] - rm[c]) * s + beta[c];
    const float v  = h[i] * s + sh;
    h[i] = v > 0.0f ? v : 0.0f;
}

// ---------------------------------------------------------------------------
// In-place bias + log_softmax per row. One wave32 per row (threadIdx.y = row
// in block); lanes cover up to 64 columns via (lane, lane+32).
// ---------------------------------------------------------------------------
__global__ __launch_bounds__(256)
void bias_log_softmax(float* __restrict__ out, const float* __restrict__ b,
                      int Nrows, int C) {
    const int row  = blockIdx.x * blockDim.y + threadIdx.y;
    const int lane = threadIdx.x;      // 0..31
    if (row >= Nrows) return;
    float* r = out + (unsigned)row * (unsigned)C;

    float v0 = -INFINITY, v1 = -INFINITY;
    if (lane < C)      v0 = r[lane]      + b[lane];
    if (lane + 32 < C) v1 = r[lane + 32] + b[lane + 32];

    float m = fmaxf(v0, v1);
    #pragma unroll
    for (int off = 16; off > 0; off >>= 1) m = fmaxf(m, __shfl_xor(m, off, 32));

    float s = 0.0f;
    if (lane < C)      s += expf(v0 - m);
    if (lane + 32 < C) s += expf(v1 - m);
    #pragma unroll
    for (int off = 16; off > 0; off >>= 1) s += __shfl_xor(s, off, 32);

    const float ls = logf(s);
    if (lane < C)      r[lane]      = v0 - m - ls;
    if (lane + 32 < C) r[lane + 32] = v1 - m - ls;
}

// ---------------------------------------------------------------------------
extern "C" void kernel_launch(void* const* d_in, const int* in_sizes, int n_in,
                              void* d_out, int out_size, void* d_ws, size_t ws_size,
                              hipStream_t stream) {
    const float* x    = (const float*)d_in[0];
    const int*   src  = (const int*)  d_in[1];
    const int*   dst  = (const int*)  d_in[2];
    const float* W1   = (const float*)d_in[3];
    const float* b1   = (const float*)d_in[4];
    const float* W2   = (const float*)d_in[5];
    const float* b2   = (const float*)d_in[6];
    const float* W3   = (const float*)d_in[7];
    const float* b3   = (const float*)d_in[8];
    const float* g1   = (const float*)d_in[9];
    const float* be1  = (const float*)d_in[10];
    const float* rm1  = (const float*)d_in[11];
    const float* rv1  = (const float*)d_in[12];
    const float* g2   = (const float*)d_in[13];
    const float* be2  = (const float*)d_in[14];
    const float* rm2  = (const float*)d_in[15];
    const float* rv2  = (const float*)d_in[16];

    const int H = 128;
    const int N = in_sizes[0] / H;     // 100000
    const int E = in_sizes[1];         // 800000
    const int C = in_sizes[8];         // 40

    float* A = (float*)d_ws;                     // N x 128 scratch (xw)
    float* B = A + (size_t)N * H;                // N x 128 scratch (agg / hidden)

    const int gemmGrid = (N + 63) / 64;
    const unsigned NH = (unsigned)N * H;
    const int zeroNHGrid = (int)((NH + 255u) / 256u);
    const unsigned aggH = (unsigned)E * (H / 4);
    const int aggHGrid = (int)((aggH + 255u) / 256u);
    const unsigned aggC = (unsigned)E * (C / 4);
    const int aggCGrid = (int)((aggC + 255u) / 256u);
    const unsigned NC = (unsigned)N * C;

    // ---- Layer 1 ----
    gemm_wmma_f32 <<<gemmGrid, 256, 0, stream>>>(x, W1, A, N, H);
    zero_f32      <<<zeroNHGrid, 256, 0, stream>>>(B, NH);
    agg_edges<32> <<<aggHGrid, 256, 0, stream>>>(A, src, dst, B, E);
    bias_bn_relu  <<<zeroNHGrid, 256, 0, stream>>>(B, b1, g1, be1, rm1, rv1, NH, H);

    // ---- Layer 2 ----
    gemm_wmma_f32 <<<gemmGrid, 256, 0, stream>>>(B, W2, A, N, H);
    zero_f32      <<<zeroNHGrid, 256, 0, stream>>>(B, NH);
    agg_edges<32> <<<aggHGrid, 256, 0, stream>>>(A, src, dst, B, E);
    bias_bn_relu  <<<zeroNHGrid, 256, 0, stream>>>(B, b2, g2, be2, rm2, rv2, NH, H);

    // ---- Layer 3 (C = 40) ----
    gemm_wmma_f32 <<<gemmGrid, 256, 0, stream>>>(B, W3, A, N, C);
    zero_f32      <<<(int)((NC + 255u) / 256u), 256, 0, stream>>>((float*)d_out, NC);
    agg_edges<10> <<<aggCGrid, 256, 0, stream>>>(A, src, dst, (float*)d_out, E);

    dim3 lsBlock(32, 8);
    bias_log_softmax <<<(N + 7) / 8, lsBlock, 0, stream>>>((float*)d_out, b3, N, C);
}